// GraphNeuralNet_57208964383047
// MI455X (gfx1250) — compile-verified
//
#include <hip/hip_runtime.h>
#include <hip/hip_bf16.h>
#include <math.h>

typedef __attribute__((ext_vector_type(2))) float v2f;
typedef __attribute__((ext_vector_type(8))) float v8f;

static constexpr int N  = 512;
static constexpr int F  = 64;
static constexpr int NN = N * N;    // 262144
static constexpr int NF = N * F;    // 32768
static constexpr int KC = 64;       // K-chunk staged in LDS
static constexpr int BTS = 66;      // transposed LDS stride (even -> b64-aligned, conflict-free)
#define BN_EPS 1e-5f

__device__ __forceinline__ v8f wmma4(v2f a, v2f b, v8f c) {
  // D = A(16x4,f32) * B(4x16,f32) + C(16x16,f32), full fp32 precision
  return __builtin_amdgcn_wmma_f32_16x16x4_f32(
      /*neg_a=*/false, a, /*neg_b=*/false, b,
      /*c_mod=*/(short)0, c, /*reuse_a=*/false, /*reuse_b=*/false);
}

// Stage a 64x64 row-major panel chunk transposed into LDS: Bt[n*BTS + k] = src[k*64 + n]
// Global reads coalesced; LDS writes conflict-free under wave32 (bank = (2n+k) & 63).
__device__ __forceinline__ void stage_bt(const float* __restrict__ src, float* __restrict__ Bt) {
#pragma unroll
  for (int i = 0; i < 16; ++i) {
    const int idx = i * 256 + threadIdx.x;
    const int k = idx >> 6, n = idx & 63;
    Bt[n * BTS + k] = src[k * 64 + n];
  }
}

__device__ __forceinline__ v2f bfrag(const float* __restrict__ Bt, int n, int kk) {
  return *reinterpret_cast<const v2f*>(Bt + n * BTS + kk);   // one ds_load_b64
}

// ---------------------------------------------------------------------------
// adjs = adjs0 + adjs1 + adjs2 (elementwise, float4)
__global__ __launch_bounds__(256)
void add3_kernel(const float* __restrict__ a, const float* __restrict__ b,
                 const float* __restrict__ c, float* __restrict__ o, int n4) {
  int i = blockIdx.x * blockDim.x + threadIdx.x;
  if (i < n4) {
    float4 x = reinterpret_cast<const float4*>(a)[i];
    float4 y = reinterpret_cast<const float4*>(b)[i];
    float4 z = reinterpret_cast<const float4*>(c)[i];
    float4 w;
    w.x = x.x + y.x + z.x; w.y = x.y + y.y + z.y;
    w.z = x.z + y.z + z.z; w.w = x.w + y.w + z.w;
    reinterpret_cast<float4*>(o)[i] = w;
  }
}

// ---------------------------------------------------------------------------
// h0 = ReLU(rowBN( hs_init[b](512x512) @ W_init(512x64) ))
__global__ __launch_bounds__(256)
void gemm_h0_kernel(const float* __restrict__ A, const float* __restrict__ W,
                    const float* __restrict__ bnMean, const float* __restrict__ bnVar,
                    const float* __restrict__ bnGamma, const float* __restrict__ bnBeta,
                    float* __restrict__ Cout) {
  __shared__ float Bt[64 * BTS];                       // 16.9 KB
  const int b     = blockIdx.y;
  const int wave  = threadIdx.x >> 5;
  const int lane  = threadIdx.x & 31;
  const int lhalf = lane >> 4, lmod = lane & 15;
  const int m0    = blockIdx.x * 128 + wave * 16;

  const float* Arow = A + (size_t)b * NN + (size_t)(m0 + lmod) * N;

  v8f acc[4];
#pragma unroll
  for (int nt = 0; nt < 4; ++nt)
#pragma unroll
    for (int r = 0; r < 8; ++r) acc[nt][r] = 0.0f;

  for (int kc = 0; kc < N; kc += KC) {
    stage_bt(W + (size_t)kc * F, Bt);
    __syncthreads();
    const float* Ak = Arow + kc;
#pragma unroll 4
    for (int ks = 0; ks < KC; ks += 4) {
      const int kk = ks + 2 * lhalf;
      v2f av = *reinterpret_cast<const v2f*>(Ak + kk);
#pragma unroll
      for (int nt = 0; nt < 4; ++nt)
        acc[nt] = wmma4(av, bfrag(Bt, nt * 16 + lmod, kk), acc[nt]);
    }
    __syncthreads();
  }

  float* Co = Cout + (size_t)b * NF;
#pragma unroll
  for (int r = 0; r < 8; ++r) {
    const int mrow = m0 + r + 8 * lhalf;
    const float s  = bnGamma[mrow] * rsqrtf(bnVar[mrow] + BN_EPS);
    const float sh = bnBeta[mrow] - bnMean[mrow] * s;
#pragma unroll
    for (int nt = 0; nt < 4; ++nt)
      Co[mrow * F + nt * 16 + lmod] = fmaxf(acc[nt][r] * s + sh, 0.0f);
  }
}

// ---------------------------------------------------------------------------
// C_j[b] = A_j[b](512x512) @ Bp[b](512x64), j=0..2. Shared B-panel staged once;
// each B fragment feeds 3 WMMAs (one per adjacency flavor).
__global__ __launch_bounds__(256)
void multiA_kernel(const float* __restrict__ A0, const float* __restrict__ A1,
                   const float* __restrict__ A2, const float* __restrict__ Bp,
                   float* __restrict__ C0, float* __restrict__ C1, float* __restrict__ C2) {
  __shared__ float Bt[64 * BTS];                       // 16.9 KB
  const int b     = blockIdx.y;
  const int wave  = threadIdx.x >> 5;
  const int lane  = threadIdx.x & 31;
  const int lhalf = lane >> 4, lmod = lane & 15;
  const int m0    = blockIdx.x * 128 + wave * 16;
  const size_t rowOff = (size_t)b * NN + (size_t)(m0 + lmod) * N;

  const float* Ar[3] = {A0 + rowOff, A1 + rowOff, A2 + rowOff};

  v8f acc[3][4];
#pragma unroll
  for (int j = 0; j < 3; ++j)
#pragma unroll
    for (int nt = 0; nt < 4; ++nt)
#pragma unroll
      for (int r = 0; r < 8; ++r) acc[j][nt][r] = 0.0f;

  for (int kc = 0; kc < N; kc += KC) {
    stage_bt(Bp + (size_t)b * NF + (size_t)kc * F, Bt);
    __syncthreads();
#pragma unroll 2
    for (int ks = 0; ks < KC; ks += 4) {
      const int kk = ks + 2 * lhalf;
      v2f a0 = *reinterpret_cast<const v2f*>(Ar[0] + kc + kk);
      v2f a1 = *reinterpret_cast<const v2f*>(Ar[1] + kc + kk);
      v2f a2 = *reinterpret_cast<const v2f*>(Ar[2] + kc + kk);
#pragma unroll
      for (int nt = 0; nt < 4; ++nt) {
        const v2f bv = bfrag(Bt, nt * 16 + lmod, kk);   // loaded once, used 3x
        acc[0][nt] = wmma4(a0, bv, acc[0][nt]);
        acc[1][nt] = wmma4(a1, bv, acc[1][nt]);
        acc[2][nt] = wmma4(a2, bv, acc[2][nt]);
      }
    }
    __syncthreads();
  }

  float* Cj[3] = {C0 + (size_t)b * NF, C1 + (size_t)b * NF, C2 + (size_t)b * NF};
#pragma unroll
  for (int j = 0; j < 3; ++j)
#pragma unroll
    for (int r = 0; r < 8; ++r) {
      const int mrow = m0 + r + 8 * lhalf;
#pragma unroll
      for (int nt = 0; nt < 4; ++nt)
        Cj[j][mrow * F + nt * 16 + lmod] = acc[j][nt][r];
    }
}

// ---------------------------------------------------------------------------
// C_j[b] += A[b](512x512) @ B_j[b](512x64), j=0..2 (C in/out). Shared A:
// one A fragment feeds 12 WMMAs; adjs is streamed once per layer.
__global__ __launch_bounds__(256)
void multiB_kernel(const float* __restrict__ A,
                   const float* __restrict__ B0, const float* __restrict__ B1,
                   const float* __restrict__ B2,
                   float* __restrict__ C0, float* __restrict__ C1, float* __restrict__ C2) {
  __shared__ float Bt[3][64 * BTS];                    // 50.7 KB
  const int b     = blockIdx.y;
  const int wave  = threadIdx.x >> 5;
  const int lane  = threadIdx.x & 31;
  const int lhalf = lane >> 4, lmod = lane & 15;
  const int m0    = blockIdx.x * 128 + wave * 16;

  const float* Arow = A + (size_t)b * NN + (size_t)(m0 + lmod) * N;
  float* Cj[3] = {C0 + (size_t)b * NF, C1 + (size_t)b * NF, C2 + (size_t)b * NF};

  v8f acc[3][4];
#pragma unroll
  for (int j = 0; j < 3; ++j)
#pragma unroll
    for (int nt = 0; nt < 4; ++nt)
#pragma unroll
      for (int r = 0; r < 8; ++r)
        acc[j][nt][r] = Cj[j][(m0 + r + 8 * lhalf) * F + nt * 16 + lmod];

  const float* Bj[3] = {B0 + (size_t)b * NF, B1 + (size_t)b * NF, B2 + (size_t)b * NF};

  for (int kc = 0; kc < N; kc += KC) {
    stage_bt(Bj[0] + (size_t)kc * F, Bt[0]);
    stage_bt(Bj[1] + (size_t)kc * F, Bt[1]);
    stage_bt(Bj[2] + (size_t)kc * F, Bt[2]);
    __syncthreads();
    const float* Ak = Arow + kc;
#pragma unroll 2
    for (int ks = 0; ks < KC; ks += 4) {
      const int kk = ks + 2 * lhalf;
      const v2f av = *reinterpret_cast<const v2f*>(Ak + kk);  // loaded once, used 12x
#pragma unroll
      for (int nt = 0; nt < 4; ++nt) {
        const int n = nt * 16 + lmod;
        acc[0][nt] = wmma4(av, bfrag(Bt[0], n, kk), acc[0][nt]);
        acc[1][nt] = wmma4(av, bfrag(Bt[1], n, kk), acc[1][nt]);
        acc[2][nt] = wmma4(av, bfrag(Bt[2], n, kk), acc[2][nt]);
      }
    }
    __syncthreads();
  }

#pragma unroll
  for (int j = 0; j < 3; ++j)
#pragma unroll
    for (int r = 0; r < 8; ++r) {
      const int mrow = m0 + r + 8 * lhalf;
#pragma unroll
      for (int nt = 0; nt < 4; ++nt)
        Cj[j][mrow * F + nt * 16 + lmod] = acc[j][nt][r];
    }
}

// ---------------------------------------------------------------------------
// Hout = ReLU(rowBN( Hc@W3l + bias + S0@W0l + S1@W1l + S2@W2l ))   (K=64 x4)
__global__ __launch_bounds__(256)
void layer_combine_kernel(const float* __restrict__ Hc,
                          const float* __restrict__ S0, const float* __restrict__ S1,
                          const float* __restrict__ S2,
                          const float* __restrict__ W3l, const float* __restrict__ W0l,
                          const float* __restrict__ W1l, const float* __restrict__ W2l,
                          const float* __restrict__ bias,
                          const float* __restrict__ bnMean, const float* __restrict__ bnVar,
                          const float* __restrict__ bnGamma, const float* __restrict__ bnBeta,
                          float* __restrict__ Hout) {
  __shared__ float Bt[64 * BTS];                       // one W at a time
  const int b     = blockIdx.y;
  const int wave  = threadIdx.x >> 5;
  const int lane  = threadIdx.x & 31;
  const int lhalf = lane >> 4, lmod = lane & 15;
  const int m0    = blockIdx.x * 128 + wave * 16;

  const float* Al[4] = {Hc, S0, S1, S2};
  const float* Bl[4] = {W3l, W0l, W1l, W2l};

  v8f acc[4];
#pragma unroll
  for (int nt = 0; nt < 4; ++nt)
#pragma unroll
    for (int r = 0; r < 8; ++r) acc[nt][r] = 0.0f;

  for (int t = 0; t < 4; ++t) {
    stage_bt(Bl[t], Bt);
    __syncthreads();
    const float* Arow = Al[t] + (size_t)b * NF + (size_t)(m0 + lmod) * F;
#pragma unroll 4
    for (int ks = 0; ks < 64; ks += 4) {
      const int kk = ks + 2 * lhalf;
      v2f av = *reinterpret_cast<const v2f*>(Arow + kk);
#pragma unroll
      for (int nt = 0; nt < 4; ++nt)
        acc[nt] = wmma4(av, bfrag(Bt, nt * 16 + lmod, kk), acc[nt]);
    }
    __syncthreads();
  }

  float* Ho = Hout + (size_t)b * NF;
#pragma unroll
  for (int r = 0; r < 8; ++r) {
    const int mrow = m0 + r + 8 * lhalf;
    const float s  = bnGamma[mrow] * rsqrtf(bnVar[mrow] + BN_EPS);
    const float sh = bnBeta[mrow] - bnMean[mrow] * s;
#pragma unroll
    for (int nt = 0; nt < 4; ++nt) {
      const int col = nt * 16 + lmod;
      Ho[mrow * F + col] = fmaxf((acc[nt][r] + bias[col]) * s + sh, 0.0f);
    }
  }
}

// ---------------------------------------------------------------------------
// graph_h[b,f] = ReLU(BN_F( sum_n h3[b,n,f] ))
__global__ __launch_bounds__(64)
void reduce_bn_kernel(const float* __restrict__ h,
                      const float* __restrict__ m, const float* __restrict__ v,
                      const float* __restrict__ g, const float* __restrict__ be,
                      float* __restrict__ out) {
  const int b = blockIdx.x;
  const int f = threadIdx.x;
  const float* p = h + (size_t)b * NF + f;
  float s = 0.0f;
  for (int n = 0; n < N; ++n) s += p[n * F];
  float y = (s - m[f]) * rsqrtf(v[f] + BN_EPS) * g[f] + be[f];
  out[b * F + f] = fmaxf(y, 0.0f);
}

// ---------------------------------------------------------------------------
// out[b,:] = softmax( graph_h[b,:] @ fc1_w.T + fc1_b )   (A=512)
__global__ __launch_bounds__(256)
void fc_softmax_kernel(const float* __restrict__ gh, const float* __restrict__ w,
                       const float* __restrict__ bias, float* __restrict__ out) {
  const int b = blockIdx.x;
  __shared__ float gsh[F];
  __shared__ float red[256];
  if (threadIdx.x < F) gsh[threadIdx.x] = gh[b * F + threadIdx.x];
  __syncthreads();

  float logit[2];
#pragma unroll
  for (int j = 0; j < 2; ++j) {
    const int a = threadIdx.x + j * 256;
    float s = bias[a];
    const float* wr = w + (size_t)a * F;
#pragma unroll 8
    for (int f = 0; f < F; ++f) s += gsh[f] * wr[f];
    logit[j] = s;
  }
  red[threadIdx.x] = fmaxf(logit[0], logit[1]);
  __syncthreads();
  for (int off = 128; off > 0; off >>= 1) {
    if (threadIdx.x < off) red[threadIdx.x] = fmaxf(red[threadIdx.x], red[threadIdx.x + off]);
    __syncthreads();
  }
  const float mx = red[0];
  __syncthreads();
  const float e0 = expf(logit[0] - mx), e1 = expf(logit[1] - mx);
  red[threadIdx.x] = e0 + e1;
  __syncthreads();
  for (int off = 128; off > 0; off >>= 1) {
    if (threadIdx.x < off) red[threadIdx.x] += red[threadIdx.x + off];
    __syncthreads();
  }
  const float inv = 1.0f / red[0];
  out[b * 512 + threadIdx.x]       = e0 * inv;
  out[b * 512 + threadIdx.x + 256] = e1 * inv;
}

// ---------------------------------------------------------------------------
extern "C" void kernel_launch(void* const* d_in, const int* in_sizes, int n_in,
                              void* d_out, int out_size, void* d_ws, size_t ws_size,
                              hipStream_t stream) {
  const float* hs_init = (const float*)d_in[0];
  const float* adj0 = (const float*)d_in[1];
  const float* adj1 = (const float*)d_in[2];
  const float* adj2 = (const float*)d_in[3];
  const float* W_init  = (const float*)d_in[4];
  const float* W0 = (const float*)d_in[5];
  const float* W1 = (const float*)d_in[6];
  const float* W2 = (const float*)d_in[7];
  const float* W3 = (const float*)d_in[8];
  const float* bvec  = (const float*)d_in[9];
  const float* fc1_w = (const float*)d_in[10];
  const float* fc1_b = (const float*)d_in[11];
  const float* bim = (const float*)d_in[12]; const float* biv = (const float*)d_in[13];
  const float* big = (const float*)d_in[14]; const float* bib = (const float*)d_in[15];
  const float* blm = (const float*)d_in[16]; const float* blv = (const float*)d_in[17];
  const float* blg = (const float*)d_in[18]; const float* blb = (const float*)d_in[19];
  const float* bgm = (const float*)d_in[20]; const float* bgv = (const float*)d_in[21];
  const float* bgg = (const float*)d_in[22]; const float* bgb = (const float*)d_in[23];

  // workspace layout (floats)
  float* ws   = (float*)d_ws;
  float* adjs = ws;                        // 16*512*512
  float* hA   = adjs + 16 * NN;            // 16*512*64 each below
  float* hB   = hA + 16 * NF;
  float* S[3] = {hB + 16 * NF, hB + 2 * 16 * NF, hB + 3 * 16 * NF};
  float* T[3] = {hB + 4 * 16 * NF, hB + 5 * 16 * NF, hB + 6 * 16 * NF};
  float* gh   = hB + 7 * 16 * NF;          // 16*64

  const dim3 gGrid(4, 16);                 // 128 rows per WG x 16 batches
  const dim3 gBlk(256);

  // adjs = adjs0 + adjs1 + adjs2
  add3_kernel<<<dim3((16 * NN / 4 + 255) / 256), gBlk, 0, stream>>>(
      adj0, adj1, adj2, adjs, 16 * NN / 4);

  // h0 = ReLU(rowBN(hs_init @ W_init))
  gemm_h0_kernel<<<gGrid, gBlk, 0, stream>>>(hs_init, W_init, bim, biv, big, bib, hA);

  float* hcur = hA;
  float* hnxt = hB;
  for (int l = 0; l < 3; ++l) {
    if (l == 0) {
      // S_k = adjs_k @ h0  (shared B-panel, fused over k)
      multiA_kernel<<<gGrid, gBlk, 0, stream>>>(adj0, adj1, adj2, hcur, S[0], S[1], S[2]);
    } else {
      // T_k = adjs_k @ h_l
      multiA_kernel<<<gGrid, gBlk, 0, stream>>>(adj0, adj1, adj2, hcur, T[0], T[1], T[2]);
      // T_k += adjs @ S_k   (shared A, fused over k)
      multiB_kernel<<<gGrid, gBlk, 0, stream>>>(adjs, S[0], S[1], S[2], T[0], T[1], T[2]);
      for (int k = 0; k < 3; ++k) { float* t = S[k]; S[k] = T[k]; T[k] = t; }
    }
    // h_{l+1} = ReLU(rowBN( h_l@W3[l] + b[l] + Σ_k S_k@W_k[l] ))
    layer_combine_kernel<<<gGrid, gBlk, 0, stream>>>(
        hcur, S[0], S[1], S[2],
        W3 + l * 4096, W0 + l * 4096, W1 + l * 4096, W2 + l * 4096,
        bvec + l * 64,
        blm + l * 512, blv + l * 512, blg + l * 512, blb + l * 512,
        hnxt);
    float* t = hcur; hcur = hnxt; hnxt = t;
  }

  // graph head
  reduce_bn_kernel<<<dim3(16), dim3(64), 0, stream>>>(hcur, bgm, bgv, bgg, bgb, gh);
  fc_softmax_kernel<<<dim3(16), gBlk, 0, stream>>>(gh, fc1_w, fc1_b, (float*)d_out);
}